// HarmonicGenAMD_25786983645325
// MI455X (gfx1250) — compile-verified
//
#include <hip/hip_runtime.h>
#include <hip/hip_bf16.h>

// ---- problem constants (match reference) ----
#define SPECIES 5
#define NRAD 3
#define NANG 3
#define NF 9
#define NFP 16          // F padded to 16 for WMMA N-dim
#define BB 8
#define MM 4096
#define NN 64
#define EPSF 1e-12f

typedef __attribute__((ext_vector_type(2))) float v2f;
typedef __attribute__((ext_vector_type(8))) float v8f;

// workspace layout (floats):
//   cu_tab : [B][N][16][4]  unit coefficient vectors (d=3 pad 0), f>=9 -> 0
//   cn_tab : [B][N][16]     |c|, f>=9 -> 0
//   par    : [B][N][3][3]   gathered (a,b,p) per radial j
#define CU_OFF  0
#define CN_OFF  (BB*NN*NFP*4)
#define PAR_OFF (CN_OFF + BB*NN*NFP)

__global__ void hg_precompute(const float* __restrict__ coef,
                              const int*   __restrict__ labels,
                              const float* __restrict__ pa,
                              const float* __restrict__ pb,
                              const float* __restrict__ pp,
                              float* __restrict__ ws) {
    int t = blockIdx.x * blockDim.x + threadIdx.x;
    if (t >= BB * NN * NFP) return;
    int f = t & 15;
    int n = (t >> 4) & (NN - 1);
    int b = t >> 10;                       // /(16*64)
    float cu0 = 0.f, cu1 = 0.f, cu2 = 0.f, cn = 0.f;
    if (f < NF) {
        const float* c = coef + ((size_t)(b * NN + n) * NF + f) * 3;
        float c0 = c[0], c1 = c[1], c2 = c[2];
        cn = sqrtf(c0 * c0 + c1 * c1 + c2 * c2);
        float inv = 1.0f / fmaxf(cn, EPSF);
        cu0 = c0 * inv; cu1 = c1 * inv; cu2 = c2 * inv;
    }
    float* cu = ws + CU_OFF + (size_t)t * 4;
    cu[0] = cu0; cu[1] = cu1; cu[2] = cu2; cu[3] = 0.f;
    ws[CN_OFF + t] = cn;
    if (f < NRAD) {                         // gather radial params via labels
        int s = labels[b * NN + n];
        float* par = ws + PAR_OFF + (size_t)(b * NN + n) * 9 + f * 3;
        par[0] = pa[s * NRAD + f];
        par[1] = pb[s * NRAD + f];
        par[2] = pp[s * NRAD + f];
    }
}

__global__ __launch_bounds__(256)
void hg_density(const float* __restrict__ dv,
                const float* __restrict__ ws,
                float* __restrict__ out) {
    __shared__ float s_cu[NN * NFP * 4];    // 16 KB
    __shared__ float s_cn[NN * NFP];        // 4 KB
    __shared__ float s_par[NN * 9];         // 2.25 KB
    __shared__ float s_stage[8][16 * 4];    // per-wave: [m16][rinv,rad0,rad1,rad2]

    const int tid  = threadIdx.x;
    const int wave = tid >> 5;
    const int lane = tid & 31;
    const int b    = blockIdx.x >> 5;       // 32 m-blocks of 128 per batch
    const int mblk = blockIdx.x & 31;
    const int m0   = mblk * 128 + wave * 16;

    // cooperative preload of per-batch atom tables into LDS
    {
        const float* gcu  = ws + CU_OFF  + (size_t)b * NN * NFP * 4;
        const float* gcn  = ws + CN_OFF  + (size_t)b * NN * NFP;
        const float* gpar = ws + PAR_OFF + (size_t)b * NN * 9;
        for (int i = tid; i < NN * NFP * 4; i += 256) s_cu[i] = gcu[i];
        for (int i = tid; i < NN * NFP;     i += 256) s_cn[i] = gcn[i];
        for (int i = tid; i < NN * 9;       i += 256) s_par[i] = gpar[i];
    }
    __syncthreads();

    const int f  = lane & 15;               // WMMA column (term index)
    const int hi = lane >> 4;               // K-half of the wave
    const int l  = (f % NANG) + 1;          // Legendre order 1..3 (cycling)
    int jt = f / NANG; if (jt >= NRAD) jt = 0;  // radial index (clamped for pad lanes)
    const int j = jt;
    // Legendre cubic coefficients: P = ((c3*x + c2)*x + c1)*x + c0
    const float c3 = (l == 3) ?  2.5f : 0.f;
    const float c2 = (l == 2) ?  1.5f : 0.f;
    const float c1 = (l == 1) ?  1.0f : ((l == 3) ? -1.5f : 0.f);
    const float c0 = (l == 2) ? -0.5f : 0.f;

    float acc[8];
#pragma unroll
    for (int i = 0; i < 8; ++i) acc[i] = 0.f;

    const int mrow = m0 + f;                // row of the A-matrix this lane feeds
    const v8f czero = {};

    for (int n = 0; n < NN; ++n) {
        // ---- A-matrix (16m x 4k f32): lanes 0-15 -> K0/K1 = dv.x/dv.y,
        //                               lanes 16-31 -> K2/K3 = dv.z/0
        const float* dvp = dv + (((size_t)b * MM + mrow) * NN + n) * 3;
        float a0, a1;
        if (hi == 0) { a0 = dvp[0]; a1 = dvp[1]; }
        else         { a0 = dvp[2]; a1 = 0.f;   }

        // r for m = m0 + (lane&15), via cross-half butterfly
        float s = a0 * a0 + a1 * a1;
        s += __shfl_xor(s, 16, 32);
        float r    = sqrtf(s);
        float rinv = 1.0f / fmaxf(r, EPSF);
        float logr = __logf(fmaxf(r, 1e-30f));

        // radial kernels a_j * r^p_j * exp(-b_j r) = a_j * exp(p_j ln r - b_j r)
        const float* par = s_par + n * 9;
        float e0 = par[0] * __expf(fmaf(par[2], logr, -par[1] * r));
        float e1 = par[3] * __expf(fmaf(par[5], logr, -par[4] * r));
        float e2 = par[6] * __expf(fmaf(par[8], logr, -par[7] * r));
        if (hi == 0) {
            float* st = &s_stage[wave][f * 4];
            st[0] = rinv; st[1] = e0; st[2] = e1; st[3] = e2;
        }
        // DS ops are in-order within a wave; fence register availability + scheduling
        asm volatile("s_wait_dscnt 0" ::: "memory");
        __builtin_amdgcn_wave_barrier();

        // ---- B-matrix (4k x 16f): column f = unit coefficient vector
        const float* cup = &s_cu[(n * NFP + f) * 4];
        v2f av, bv;
        av.x = a0; av.y = a1;
        bv.x = cup[hi ? 2 : 0];
        bv.y = hi ? 0.f : cup[1];

        // D[m,f] = dv . c_hat  (K=3 padded to 4)
        v8f dmat = __builtin_amdgcn_wmma_f32_16x16x4_f32(
            false, av, false, bv, (short)0, czero, false, false);

        const float cnv = s_cn[n * NFP + f];
#pragma unroll
        for (int i = 0; i < 8; ++i) {
            const float* st = &s_stage[wave][(i + 8 * hi) * 4];
            float rinv_i = st[0];
            float rad_i  = st[1 + j];
            float cosv = dmat[i] * rinv_i;
            float P = fmaf(fmaf(fmaf(c3, cosv, c2), cosv, c1), cosv, c0);
            acc[i] = fmaf(rad_i * cnv, P, acc[i]);
        }
        asm volatile("" ::: "memory");  // keep next-iter stage stores below these reads
    }

    // reduce over the 16 f-lanes within each half (f>=9 lanes contribute 0)
#pragma unroll
    for (int i = 0; i < 8; ++i) {
        acc[i] += __shfl_xor(acc[i], 1, 32);
        acc[i] += __shfl_xor(acc[i], 2, 32);
        acc[i] += __shfl_xor(acc[i], 4, 32);
        acc[i] += __shfl_xor(acc[i], 8, 32);
    }
    // lane with (lane&15)==i (one per half) writes out[b, m0 + i + 8*hi]
#pragma unroll
    for (int i = 0; i < 8; ++i) {
        if ((lane & 15) == i) {
            out[(size_t)b * MM + m0 + i + 8 * hi] = acc[i];
        }
    }
}

extern "C" void kernel_launch(void* const* d_in, const int* in_sizes, int n_in,
                              void* d_out, int out_size, void* d_ws, size_t ws_size,
                              hipStream_t stream) {
    const float* dv     = (const float*)d_in[0];
    const float* coef   = (const float*)d_in[1];
    const int*   labels = (const int*)  d_in[2];
    const float* pa     = (const float*)d_in[3];
    const float* pb     = (const float*)d_in[4];
    const float* pp     = (const float*)d_in[5];
    float* out = (float*)d_out;
    float* ws  = (float*)d_ws;

    // precompute atom tables: B*N*16 = 8192 threads
    hg_precompute<<<(BB * NN * NFP + 255) / 256, 256, 0, stream>>>(
        coef, labels, pa, pb, pp, ws);

    // main: 8 waves/block, each wave owns a 16-m tile; B * M/128 = 256 blocks
    hg_density<<<BB * (MM / 128), 256, 0, stream>>>(dv, ws, out);
}